// HeteGCNLayer_38173669327309
// MI455X (gfx1250) — compile-verified
//
#include <hip/hip_runtime.h>

typedef __attribute__((ext_vector_type(2))) float v2f;
typedef __attribute__((ext_vector_type(4))) float v4f;
typedef __attribute__((ext_vector_type(8))) float v8f;

#define DOUT 256
#define DIN  512
#define N_P  20000
#define N_A  10000
#define N_T  5000

// ---------------------------------------------------------------- utilities
__global__ __launch_bounds__(256) void zero_f32(float* __restrict__ p, long long count) {
    long long i = (long long)blockIdx.x * 256 + threadIdx.x;
    if (i < count) p[i] = 0.0f;
}

// ---------------------------------------------------------------- f32 WMMA GEMM
// C[M,256] = A[M,K] @ B[K,256]  (+ C if accum) (+ bias[col] if bias != null)
// Block: 256 threads (8 waves). Block tile 128x64, BK = 32.
// Per wave: 16 rows x 64 cols = 4 accumulators of v8f, V_WMMA_F32_16X16X4_F32.
__global__ __launch_bounds__(256) void gemm_f32_wmma(
    const float* __restrict__ A, const float* __restrict__ B,
    float* __restrict__ C, int M, int K,
    const float* __restrict__ bias, int accum) {
    __shared__ float Bs[32][64];

    const int tid  = threadIdx.x;
    const int w    = tid >> 5;       // wave id (wave32)
    const int lane = tid & 31;
    const int lr   = lane & 15;      // lane row / col within fragment
    const int hi   = lane >> 4;      // which half of the wave
    const int rowBase = blockIdx.y * 128 + w * 16;
    const int colBase = blockIdx.x * 64;

    v8f acc[4];
#pragma unroll
    for (int i = 0; i < 4; ++i) acc[i] = (v8f)0.0f;

    // Clamp the A row: garbage in A row m only affects D row m, which is
    // never stored when m >= M. Clamping (instead of predication) keeps the
    // k-loop free of EXEC save/restore and branches.
    int arow = rowBase + lr;
    if (arow >= M) arow = M - 1;
    const float* Arow = A + (size_t)arow * K;

    // B staging: 32x64 tile = 512 float4; 256 threads x 2 float4 each.
    const int sr0 = tid >> 4;          // rows 0..15  (f4 index tid:   row = tid/16)
    const int sc0 = (tid & 15) * 4;    // col 0..60 step 4
    const int sr1 = sr0 + 16;          // rows 16..31 (f4 index tid+256)

    for (int k0 = 0; k0 < K; k0 += 32) {
        const float* Bt = B + (size_t)k0 * DOUT + colBase;
        *(v4f*)&Bs[sr0][sc0] = *(const v4f*)(Bt + (size_t)sr0 * DOUT + sc0);
        *(v4f*)&Bs[sr1][sc0] = *(const v4f*)(Bt + (size_t)sr1 * DOUT + sc0);
        __syncthreads();

        const float* Ap = Arow + k0 + 2 * hi;
#pragma unroll
        for (int kk = 0; kk < 8; ++kk) {
            // A fragment: lanes 0-15 -> K = kk*4+{0,1}; lanes 16-31 -> K = kk*4+{2,3}
            v2f a; a.x = Ap[kk * 4]; a.y = Ap[kk * 4 + 1];
#pragma unroll
            for (int nt = 0; nt < 4; ++nt) {
                // B fragment: VGPR0 = rows K{0,2}, VGPR1 = rows K{1,3}, lanes = N
                v2f b;
                b.x = Bs[kk * 4 + 2 * hi    ][nt * 16 + lr];
                b.y = Bs[kk * 4 + 2 * hi + 1][nt * 16 + lr];
                acc[nt] = __builtin_amdgcn_wmma_f32_16x16x4_f32(
                    false, a, false, b, (short)0, acc[nt], false, false);
            }
        }
        __syncthreads();
    }

    // D layout: VGPR r <-> row (r + 8*hi), lane lr <-> col
#pragma unroll
    for (int nt = 0; nt < 4; ++nt) {
        int col = colBase + nt * 16 + lr;
#pragma unroll
        for (int r = 0; r < 8; ++r) {
            int row = rowBase + r + 8 * hi;
            if (row < M) {
                float v = acc[nt][r];
                size_t idx = (size_t)row * DOUT + col;
                if (accum) v += C[idx];
                if (bias)  v += bias[col];
                C[idx] = v;
            }
        }
    }
}

// ---------------------------------------------------------------- SpMM scatter
// S[dst[e], :] += ew[e] * P[src[e], :]   (one block per edge, 256 lanes = row)
__global__ __launch_bounds__(256) void spmm_scatter(
    const float* __restrict__ P, const int* __restrict__ src,
    const int* __restrict__ dst, const float* __restrict__ ew,
    float* __restrict__ S, int E) {
    int e = blockIdx.x;
    if (e >= E) return;
    int f = threadIdx.x;
    int s = src[e], d = dst[e];
    float wv = ew[e];
    atomicAdd(&S[(size_t)d * DOUT + f], wv * P[(size_t)s * DOUT + f]);
}

// ---------------------------------------------------------------- attention
// e01[r]   = leaky_relu(nb1[r].watt[0:256] + self[r].watt[256:512])
// e01[n+r] = leaky_relu(nb2[r].watt[0:256] + self[r].watt[256:512])
__global__ __launch_bounds__(256) void att_logits(
    const float* __restrict__ nb1, const float* __restrict__ nb2,
    const float* __restrict__ selfft, const float* __restrict__ watt,
    float* __restrict__ e01, int n) {
    int r = blockIdx.x * 256 + threadIdx.x;
    if (r >= n) return;
    const float* p1 = nb1    + (size_t)r * DOUT;
    const float* p2 = nb2    + (size_t)r * DOUT;
    const float* ps = selfft + (size_t)r * DOUT;
    float s1 = 0.f, s2 = 0.f, ss = 0.f;
    for (int j = 0; j < DOUT; ++j) {
        float wn = watt[j], wsf = watt[DOUT + j];
        s1 += p1[j] * wn;
        s2 += p2[j] * wn;
        ss += ps[j] * wsf;
    }
    float l1 = s1 + ss, l2 = s2 + ss;
    l1 = l1 > 0.f ? l1 : 0.01f * l1;   // jax.nn.leaky_relu default slope
    l2 = l2 > 0.f ? l2 : 0.01f * l2;
    e01[r]     = l1;
    e01[n + r] = l2;
}

// Faithful e.view(-1,2) softmax: row r uses eflat[2r], eflat[2r+1] where
// eflat = [e0 ; e1] (exactly how e01 is laid out). agg written into nb1.
__global__ __launch_bounds__(256) void att_combine(
    float* __restrict__ nb1, const float* __restrict__ nb2,
    const float* __restrict__ e01, int n) {
    long long t = (long long)blockIdx.x * 256 + threadIdx.x;
    if (t >= (long long)n * DOUT) return;
    int r = (int)(t >> 8);
    float l0 = e01[2 * r], l1 = e01[2 * r + 1];
    float m  = fmaxf(l0, l1);
    float a0 = expf(l0 - m), a1 = expf(l1 - m);
    float att0 = a0 / (a0 + a1);
    nb1[t] = att0 * nb1[t] + (1.0f - att0) * nb2[t];
}

// ---------------------------------------------------------------- host side
namespace {

struct Rel {
    const float* x; const float* W;
    const int* src; const int* dst; const float* ew;
    int m; int E;
};

inline void gemm(const float* A, const float* B, float* C, int M, int K,
                 const float* bias, int accum, hipStream_t s) {
    dim3 grid(DOUT / 64, (M + 127) / 128);
    gemm_f32_wmma<<<grid, 256, 0, s>>>(A, B, C, M, K, bias, accum);
}

inline void run_target(const float* x_self, int n,
                       const float* ws_k, const float* wsh, const float* watt,
                       const float* wcat, const float* bk,
                       Rel r1, Rel r2, float* outp,
                       float* bA, float* bB, float* bC, float* bD, float* bE,
                       float* e01, hipStream_t s) {
    const long long cnt = (long long)n * DOUT;
    const unsigned  zb  = (unsigned)((cnt + 255) / 256);

    // self_ft = (x @ w_self) @ w_share
    gemm(x_self, ws_k, bA, n, DIN, nullptr, 0, s);
    gemm(bA, wsh, bB, n, DOUT, nullptr, 0, s);

    // relation 1: project, scatter (segment_sum), @ w_share
    gemm(r1.x, r1.W, bA, r1.m, DIN, nullptr, 0, s);
    zero_f32<<<zb, 256, 0, s>>>(bC, cnt);
    spmm_scatter<<<r1.E, 256, 0, s>>>(bA, r1.src, r1.dst, r1.ew, bC, r1.E);
    gemm(bC, wsh, bD, n, DOUT, nullptr, 0, s);

    // relation 2
    gemm(r2.x, r2.W, bA, r2.m, DIN, nullptr, 0, s);
    zero_f32<<<zb, 256, 0, s>>>(bC, cnt);
    spmm_scatter<<<r2.E, 256, 0, s>>>(bA, r2.src, r2.dst, r2.ew, bC, r2.E);
    gemm(bC, wsh, bE, n, DOUT, nullptr, 0, s);

    // attention (faithful scrambled reshape semantics) + combine into bD
    att_logits<<<(n + 255) / 256, 256, 0, s>>>(bD, bE, bB, watt, e01, n);
    att_combine<<<zb, 256, 0, s>>>(bD, bE, e01, n);

    // out = agg @ wcat[:256] + self_ft @ wcat[256:] + b
    gemm(bD, wcat,               outp, n, DOUT, nullptr, 0, s);
    gemm(bB, wcat + DOUT * DOUT, outp, n, DOUT, bk, 1, s);
}

} // namespace

extern "C" void kernel_launch(void* const* d_in, const int* in_sizes, int n_in,
                              void* d_out, int out_size, void* d_ws, size_t ws_size,
                              hipStream_t stream) {
    (void)n_in; (void)out_size; (void)ws_size;

    const float* x_p = (const float*)d_in[0];
    const float* x_a = (const float*)d_in[1];
    const float* x_t = (const float*)d_in[2];

    auto F = [&](int i) { return (const float*)d_in[i]; };
    auto I = [&](int i) { return (const int*)d_in[i]; };

    float* out = (float*)d_out;
    float* ws  = (float*)d_ws;
    const size_t cap = (size_t)N_P * DOUT;     // largest row count is N_P
    float* bA  = ws;
    float* bB  = ws + cap;
    float* bC  = ws + 2 * cap;
    float* bD  = ws + 3 * cap;
    float* bE  = ws + 4 * cap;
    float* e01 = ws + 5 * cap;                 // 2 * N_P floats

    // target p: neighbors a (edges 3..5), t (edges 6..8); params 21..27
    run_target(x_p, N_P, F(23), F(24), F(25), F(26), F(27),
               Rel{x_a, F(21), I(3), I(4), F(5), N_A, in_sizes[3]},
               Rel{x_t, F(22), I(6), I(7), F(8), N_T, in_sizes[6]},
               out, bA, bB, bC, bD, bE, e01, stream);

    // target a: neighbors p (9..11), t (12..14); params 28..34
    run_target(x_a, N_A, F(30), F(31), F(32), F(33), F(34),
               Rel{x_p, F(28), I(9),  I(10), F(11), N_P, in_sizes[9]},
               Rel{x_t, F(29), I(12), I(13), F(14), N_T, in_sizes[12]},
               out + (size_t)N_P * DOUT, bA, bB, bC, bD, bE, e01, stream);

    // target t: neighbors p (15..17), a (18..20); params 35..41
    run_target(x_t, N_T, F(37), F(38), F(39), F(40), F(41),
               Rel{x_p, F(35), I(15), I(16), F(17), N_P, in_sizes[15]},
               Rel{x_a, F(36), I(18), I(19), F(20), N_A, in_sizes[18]},
               out + (size_t)(N_P + N_A) * DOUT, bA, bB, bC, bD, bE, e01, stream);
}